// HeteroGNN_28192165331399
// MI455X (gfx1250) — compile-verified
//
#include <hip/hip_runtime.h>
#include <hip/hip_bf16.h>

// ---------------------------------------------------------------------------
// HeteroGNN on MI455X (gfx1250, wave32).
//   Phase 1: edge scatter — non-returning global_atomic_add_f32 segment-sum
//            of D=3 features + count; 4 edges/thread via b128 NT loads.
//   Phase 2: per-node fused SAGE via V_WMMA_F32_16X16X4_F32 ([N,12]@[12,64])
// ---------------------------------------------------------------------------

#define NG 20000
#define NC 200000
#define EGG 1000000
#define ECC 2000000
#define ECG 8000000
#define EGC 8000000
#define HID 64

typedef float v2f __attribute__((ext_vector_type(2)));
typedef float v8f __attribute__((ext_vector_type(8)));
typedef int   v4i __attribute__((ext_vector_type(4)));

// Non-returning relaxed agent-scope f32 atomic add -> global_atomic_add_f32
// (no CAS loop, no return data; tracked with STOREcnt).
__device__ __forceinline__ void atomic_add_f32(float* p, float v) {
    (void)__hip_atomic_fetch_add(p, v, __ATOMIC_RELAXED, __HIP_MEMORY_SCOPE_AGENT);
}

// ---------------------------------------------------------------------------
__global__ __launch_bounds__(256) void zero_f32(float* __restrict__ p, int n) {
    int i = blockIdx.x * blockDim.x + threadIdx.x;
    if (i < n) p[i] = 0.0f;
}

// ---------------------------------------------------------------------------
// 4 edges per thread: acc[dst] += (x_src, 1).  Edge stream is NT-hinted b128
// so 152MB of indices do not evict the 7MB of accumulators from L2.
// E is a multiple of 4 for every edge type in this problem.
__global__ __launch_bounds__(256) void scatter_accum4(
    const int* __restrict__ ei,     // [2, E] : row0 = src, row1 = dst
    int E,
    const float* __restrict__ xsrc, // [n_src, 3]
    float* __restrict__ acc)        // [n_dst, 4] = (sx, sy, sz, cnt)
{
    int q = blockIdx.x * blockDim.x + threadIdx.x;   // quad-edge index
    int nQuads = E >> 2;
    if (q >= nQuads) return;

    const v4i* eiS = (const v4i*)ei;                  // sources, quads
    const v4i* eiD = (const v4i*)(ei + E);            // dests, quads
    v4i s4 = __builtin_nontemporal_load(&eiS[q]);
    v4i d4 = __builtin_nontemporal_load(&eiD[q]);

#pragma unroll
    for (int k = 0; k < 4; ++k) {
        int src = s4[k];
        int dst = d4[k];
        float x0 = xsrc[src * 3 + 0];
        float x1 = xsrc[src * 3 + 1];
        float x2 = xsrc[src * 3 + 2];
        float* a = acc + (size_t)dst * 4;
        atomic_add_f32(a + 0, x0);
        atomic_add_f32(a + 1, x1);
        atomic_add_f32(a + 2, x2);
        atomic_add_f32(a + 3, 1.0f);
    }
}

// ---------------------------------------------------------------------------
// Per-node fused SAGE + ReLU + output linear for one destination node type.
// Each wave owns a 16-node tile:  A[16x12] @ Wc[12x64] via 12 x WMMA f32
// 16x16x4 (3 K-steps x 4 N-tiles), relu, then [16x64]@[64x2] scalar finish.
// nNodes is a multiple of 16, so every live wave has a full tile and EXEC is
// all-ones at every WMMA (ISA requirement).
#define WAVES_PER_BLOCK 8

__global__ __launch_bounds__(256) void node_forward(
    int nNodes,
    const float* __restrict__ x,     // [n,3] dest-type features
    const float* __restrict__ accA,  // [n,4] segment sums, edge type A
    const float* __restrict__ accB,  // [n,4] segment sums, edge type B
    const float* __restrict__ WlA, const float* __restrict__ blA, const float* __restrict__ WrA,
    const float* __restrict__ WlB, const float* __restrict__ blB, const float* __restrict__ WrB,
    const float* __restrict__ Wout,  // [64,2]
    const float* __restrict__ bout,  // [2]
    float* __restrict__ out)         // [n,2]
{
    __shared__ float sWc[12 * 64];                 // combined [12 x 64] weights
    __shared__ float sWout[64 * 2];
    __shared__ float sBout[2];
    __shared__ float sA[WAVES_PER_BLOCK][16][12];  // per-wave A tile
    __shared__ float sH[WAVES_PER_BLOCK][16][65];  // per-wave relu(h), padded row

    const int tid = threadIdx.x;

    // Build combined weight matrix:
    //   rows 0-2 : 0.5*WlA   rows 3-5 : 0.5*WlB
    //   rows 6-8 : 0.5*(WrA+WrB)   row 9 : 0.5*(blA+blB)   rows 10-11 : 0
    for (int i = tid; i < 12 * 64; i += blockDim.x) {
        int r = i >> 6, c = i & 63;
        float v;
        if (r < 3)        v = 0.5f * WlA[r * 64 + c];
        else if (r < 6)   v = 0.5f * WlB[(r - 3) * 64 + c];
        else if (r < 9)   v = 0.5f * (WrA[(r - 6) * 64 + c] + WrB[(r - 6) * 64 + c]);
        else if (r == 9)  v = 0.5f * (blA[c] + blB[c]);
        else              v = 0.0f;
        sWc[i] = v;
    }
    for (int i = tid; i < 128; i += blockDim.x) sWout[i] = Wout[i];
    if (tid < 2) sBout[tid] = bout[tid];
    __syncthreads();

    const int wave = tid >> 5;
    const int lane = tid & 31;
    const int nTiles = nNodes >> 4;                      // exact (nNodes % 16 == 0)
    const int tile = blockIdx.x * WAVES_PER_BLOCK + wave;
    if (tile >= nTiles) return;                          // whole-wave uniform exit
    const int base = tile << 4;

    // ---- stage A tile: lane m (<16) builds row m = [meanA, meanB, x, 1, 0, 0]
    if (lane < 16) {
        int node = base + lane;
        const float* aA = accA + (size_t)node * 4;
        const float* aB = accB + (size_t)node * 4;
        float cA = fmaxf(aA[3], 1.0f);
        float cB = fmaxf(aB[3], 1.0f);
        float* row = &sA[wave][lane][0];
        row[0] = aA[0] / cA;  row[1] = aA[1] / cA;  row[2] = aA[2] / cA;
        row[3] = aB[0] / cB;  row[4] = aB[1] / cB;  row[5] = aB[2] / cB;
        row[6] = x[node * 3 + 0]; row[7] = x[node * 3 + 1]; row[8] = x[node * 3 + 2];
        row[9] = 1.0f; row[10] = 0.0f; row[11] = 0.0f;
    }
    // wave-local visibility: wait for this wave's LDS stores (DScnt == 0)
    asm volatile("s_wait_dscnt 0" ::: "memory");

    // ---- gather WMMA fragments per ISA layout
    const int m     = lane & 15;
    const int khalf = (lane >> 4) << 1;   // lanes 0-15 -> K{0,1}; 16-31 -> K{2,3}
    const int mrow0 = (lane >> 4) << 3;   // C/D: lanes 0-15 rows 0-7; 16-31 rows 8-15

    v2f Af[3];
#pragma unroll
    for (int kc = 0; kc < 3; ++kc) {
        Af[kc].x = sA[wave][m][kc * 4 + khalf + 0];
        Af[kc].y = sA[wave][m][kc * 4 + khalf + 1];
    }

#pragma unroll
    for (int nt = 0; nt < 4; ++nt) {
        const int n = nt * 16 + m;
        v8f acc = {};
#pragma unroll
        for (int kc = 0; kc < 3; ++kc) {
            v2f Bf;
            Bf.x = sWc[(kc * 4 + khalf + 0) * 64 + n];
            Bf.y = sWc[(kc * 4 + khalf + 1) * 64 + n];
            acc = __builtin_amdgcn_wmma_f32_16x16x4_f32(
                false, Af[kc], false, Bf, (short)0, acc, false, false);
        }
        // relu + stash h tile in LDS (row-padded to 65 -> conflict-free reads)
#pragma unroll
        for (int v = 0; v < 8; ++v) {
            sH[wave][mrow0 + v][n] = fmaxf(acc[v], 0.0f);
        }
    }
    asm volatile("s_wait_dscnt 0" ::: "memory");

    // ---- output linear: lane m finishes node (base+m): [64]@[64x2] + b
    if (lane < 16) {
        int node = base + lane;
        float o0 = sBout[0], o1 = sBout[1];
#pragma unroll 8
        for (int n = 0; n < 64; ++n) {
            float h = sH[wave][lane][n];
            o0 += h * sWout[n * 2 + 0];
            o1 += h * sWout[n * 2 + 1];
        }
        out[(size_t)node * 2 + 0] = o0;
        out[(size_t)node * 2 + 1] = o1;
    }
}

// ---------------------------------------------------------------------------
extern "C" void kernel_launch(void* const* d_in, const int* in_sizes, int n_in,
                              void* d_out, int out_size, void* d_ws, size_t ws_size,
                              hipStream_t stream) {
    const float* x_gene = (const float*)d_in[0];
    const float* x_cell = (const float*)d_in[1];
    const int*   ei_gg  = (const int*)d_in[2];
    const int*   ei_cc  = (const int*)d_in[3];
    const int*   ei_cg  = (const int*)d_in[4];
    const int*   ei_gc  = (const int*)d_in[5];
    const float* Wl_gg = (const float*)d_in[6],  *bl_gg = (const float*)d_in[7],  *Wr_gg = (const float*)d_in[8];
    const float* Wl_cc = (const float*)d_in[9],  *bl_cc = (const float*)d_in[10], *Wr_cc = (const float*)d_in[11];
    const float* Wl_cg = (const float*)d_in[12], *bl_cg = (const float*)d_in[13], *Wr_cg = (const float*)d_in[14];
    const float* Wl_gc = (const float*)d_in[15], *bl_gc = (const float*)d_in[16], *Wr_gc = (const float*)d_in[17];
    const float* W_gene = (const float*)d_in[18], *b_gene = (const float*)d_in[19];
    const float* W_cell = (const float*)d_in[20], *b_cell = (const float*)d_in[21];

    // Workspace: four (sum3, cnt) accumulator arrays, 1.76M floats (~7MB).
    float* acc_gg = (float*)d_ws;             // [NG*4]
    float* acc_cg = acc_gg + (size_t)NG * 4;  // [NG*4]
    float* acc_cc = acc_cg + (size_t)NG * 4;  // [NC*4]
    float* acc_gc = acc_cc + (size_t)NC * 4;  // [NC*4]
    const int nAcc = 2 * NG * 4 + 2 * NC * 4;

    float* out_gene = (float*)d_out;                      // [NG,2]
    float* out_cell = (float*)d_out + (size_t)NG * 2;     // [NC,2]

    // 1) zero accumulators
    zero_f32<<<(nAcc + 255) / 256, 256, 0, stream>>>((float*)d_ws, nAcc);

    // 2) edge scatters (non-returning f32 atomics; accumulators L2-resident)
    scatter_accum4<<<(EGG / 4 + 255) / 256, 256, 0, stream>>>(ei_gg, EGG, x_gene, acc_gg);
    scatter_accum4<<<(ECG / 4 + 255) / 256, 256, 0, stream>>>(ei_cg, ECG, x_cell, acc_cg);
    scatter_accum4<<<(ECC / 4 + 255) / 256, 256, 0, stream>>>(ei_cc, ECC, x_cell, acc_cc);
    scatter_accum4<<<(EGC / 4 + 255) / 256, 256, 0, stream>>>(ei_gc, EGC, x_gene, acc_gc);

    // 3) node-level fused SAGE + output head (WMMA)
    {
        int tiles = NG / 16;                                    // 1250
        int blocks = (tiles + WAVES_PER_BLOCK - 1) / WAVES_PER_BLOCK;
        node_forward<<<blocks, 256, 0, stream>>>(
            NG, x_gene, acc_gg, acc_cg,
            Wl_gg, bl_gg, Wr_gg, Wl_cg, bl_cg, Wr_cg,
            W_gene, b_gene, out_gene);
    }
    {
        int tiles = NC / 16;                                    // 12500
        int blocks = (tiles + WAVES_PER_BLOCK - 1) / WAVES_PER_BLOCK;
        node_forward<<<blocks, 256, 0, stream>>>(
            NC, x_cell, acc_cc, acc_gc,
            Wl_cc, bl_cc, Wr_cc, Wl_gc, bl_gc, Wr_gc,
            W_cell, b_cell, out_cell);
    }
}